// DenseGCNLayer_2937757631000
// MI455X (gfx1250) — compile-verified
//
#include <hip/hip_runtime.h>

// ---------------------------------------------------------------------------
// DenseGCNLayer for MI455X (gfx1250):
//   h = x @ W^T           -> V_WMMA_F32_16X16X4_F32 (exact f32, matches ref)
//   GCN normalize/scatter -> native global_atomic_add_f32, L2-resident
//   BatchNorm + ReLU      -> column-parallel streaming passes
// N = 50000 (= 3125*16), D_IN = D_OUT = 256, E = 800000.
// ---------------------------------------------------------------------------

typedef __attribute__((ext_vector_type(2))) float v2f;
typedef __attribute__((ext_vector_type(4))) float v4f;
typedef __attribute__((ext_vector_type(8))) float v8f;

static constexpr int DF = 256;          // feature dim (in == out)
static constexpr float BN_EPS_F = 1e-5f;

// ---- degree / normalization -----------------------------------------------

__global__ void k_init_deg(float* __restrict__ deg, int N) {
  int i = blockIdx.x * blockDim.x + threadIdx.x;
  if (i < N) deg[i] = 1.0f;             // self-loop contributes 1
}

__global__ void k_count_deg(const int* __restrict__ col, float* __restrict__ deg, int E) {
  int e = blockIdx.x * blockDim.x + threadIdx.x;
  if (e < E) unsafeAtomicAdd(&deg[col[e]], 1.0f);   // global_atomic_add_f32
}

__global__ void k_deg_to_dinv(float* __restrict__ deg, int N) {
  int i = blockIdx.x * blockDim.x + threadIdx.x;
  if (i < N) deg[i] = rsqrtf(deg[i]);   // deg >= 1 always (self-loops)
}

// ---- GEMM: h[n,o] = sum_k x[n,k] * W[o,k]  (f32 WMMA 16x16x4) -------------
// Block = 128 threads = 4 waves. Block owns a 16-row tile; wave w owns the
// 64-column strip [w*64, w*64+64). 64 k-steps of K=4; A reused across 4 WMMAs.
// A layout: lane<16 -> M=lane, K={ks*4+0,1}; lane>=16 -> K={ks*4+2,3} (float2)
// B layout (mirror): lane%16 = N, lane/16 selects K pair; B[k][n] = W[n*K+k].

__global__ __launch_bounds__(128)
void k_gemm_wmma(const float* __restrict__ x, const float* __restrict__ W,
                 float* __restrict__ h, int N) {
  const int lane   = threadIdx.x & 31;
  const int wave   = threadIdx.x >> 5;        // 0..3
  const int m      = lane & 15;
  const int khi    = lane >> 4;               // 0 or 1 -> K offset 0 / 2
  const int rowbase = blockIdx.x * 16;
  const int colbase = wave * 64;

  int rA = rowbase + m;
  if (rA >= N) rA = N - 1;                    // clamp loads; stores are guarded
  const float* __restrict__ xrow = x + (size_t)rA * DF;

  const float* __restrict__ w0 = W + (size_t)(colbase +  0 + m) * DF;
  const float* __restrict__ w1 = W + (size_t)(colbase + 16 + m) * DF;
  const float* __restrict__ w2 = W + (size_t)(colbase + 32 + m) * DF;
  const float* __restrict__ w3 = W + (size_t)(colbase + 48 + m) * DF;

  v8f acc0 = {}, acc1 = {}, acc2 = {}, acc3 = {};

#pragma unroll 4
  for (int ks = 0; ks < DF / 4; ++ks) {
    const int kA = ks * 4 + khi * 2;
    v2f a  = *(const v2f*)(xrow + kA);
    v2f b0 = *(const v2f*)(w0 + kA);
    v2f b1 = *(const v2f*)(w1 + kA);
    v2f b2 = *(const v2f*)(w2 + kA);
    v2f b3 = *(const v2f*)(w3 + kA);
    acc0 = __builtin_amdgcn_wmma_f32_16x16x4_f32(false, a, false, b0, (short)0, acc0, false, false);
    acc1 = __builtin_amdgcn_wmma_f32_16x16x4_f32(false, a, false, b1, (short)0, acc1, false, false);
    acc2 = __builtin_amdgcn_wmma_f32_16x16x4_f32(false, a, false, b2, (short)0, acc2, false, false);
    acc3 = __builtin_amdgcn_wmma_f32_16x16x4_f32(false, a, false, b3, (short)0, acc3, false, false);
  }

  // C/D layout: VGPR j holds M = j + 8*khi, N = lane%16
  const int n = lane & 15;
#pragma unroll
  for (int j = 0; j < 8; ++j) {
    const int r = rowbase + j + 8 * khi;
    if (r < N) {
      float* __restrict__ hr = h + (size_t)r * DF + colbase + n;
      hr[ 0] = acc0[j];
      hr[16] = acc1[j];
      hr[32] = acc2[j];
      hr[48] = acc3[j];
    }
  }
}

// ---- agg init: self-loop message + bias -----------------------------------
// agg[r,c] = h[r,c]*dinv[r]^2 + bias[c]

__global__ __launch_bounds__(256)
void k_agg_init(const float* __restrict__ h, const float* __restrict__ dinv,
                const float* __restrict__ bias, float* __restrict__ agg, int N) {
  const int c = threadIdx.x;
  const int r = blockIdx.x;
  const float d  = dinv[r];
  const size_t i = (size_t)r * DF + c;
  agg[i] = h[i] * d * d + bias[c];
}

// ---- edge scatter: one wave per edge, grid-stride -------------------------
// lane covers 8 contiguous floats -> wave reads/writes a coalesced 1KB row.

__global__ __launch_bounds__(256)
void k_scatter(const int* __restrict__ row, const int* __restrict__ col,
               const float* __restrict__ dinv, const float* __restrict__ h,
               float* __restrict__ agg, int E) {
  const int lane = threadIdx.x & 31;
  const int wid  = (blockIdx.x * blockDim.x + threadIdx.x) >> 5;
  const int nw   = (gridDim.x * blockDim.x) >> 5;

  for (int e = wid; e < E; e += nw) {
    const int r = row[e];
    const int c = col[e];
    const float w = dinv[r] * dinv[c];
    const float* __restrict__ src = h   + (size_t)r * DF + lane * 8;
    float*       __restrict__ dst = agg + (size_t)c * DF + lane * 8;
    v4f p0 = *(const v4f*)(src);
    v4f p1 = *(const v4f*)(src + 4);
#pragma unroll
    for (int j = 0; j < 4; ++j) unsafeAtomicAdd(dst + j,     p0[j] * w);
#pragma unroll
    for (int j = 0; j < 4; ++j) unsafeAtomicAdd(dst + 4 + j, p1[j] * w);
  }
}

// ---- BatchNorm stats: per-column sum / sumsq ------------------------------

__global__ void k_zero_stats(float* __restrict__ stats) {
  stats[threadIdx.x] = 0.0f;                       // <<<1, 2*DF>>>
}

__global__ __launch_bounds__(256)
void k_stats(const float* __restrict__ agg, float* __restrict__ stats, int N) {
  const int c  = threadIdx.x;                      // column
  const int r0 = blockIdx.x * 128;
  const int r1 = min(r0 + 128, N);
  float s = 0.0f, s2 = 0.0f;
  for (int r = r0; r < r1; ++r) {
    const float v = agg[(size_t)r * DF + c];       // coalesced across threads
    s += v;
    s2 += v * v;
  }
  unsafeAtomicAdd(&stats[c],      s);
  unsafeAtomicAdd(&stats[DF + c], s2);
}

// ---- BN apply + ReLU ------------------------------------------------------

__global__ __launch_bounds__(256)
void k_bn_relu(const float* __restrict__ agg, const float* __restrict__ stats,
               const float* __restrict__ gamma, const float* __restrict__ beta,
               float* __restrict__ out, int N) {
  const int c = threadIdx.x;
  const int r = blockIdx.x;
  const float invN  = 1.0f / (float)N;
  const float mean  = stats[c] * invN;
  const float var   = stats[DF + c] * invN - mean * mean;   // biased var
  const float scale = gamma[c] * rsqrtf(var + BN_EPS_F);
  const float shift = beta[c] - mean * scale;
  const size_t i = (size_t)r * DF + c;
  const float v = agg[i] * scale + shift;
  out[i] = fmaxf(v, 0.0f);
}

// ---------------------------------------------------------------------------

extern "C" void kernel_launch(void* const* d_in, const int* in_sizes, int n_in,
                              void* d_out, int out_size, void* d_ws, size_t ws_size,
                              hipStream_t stream) {
  const float* x     = (const float*)d_in[0];
  const int*   ei    = (const int*)d_in[1];     // [2, E] flat: row then col
  const float* W     = (const float*)d_in[2];
  const float* bias  = (const float*)d_in[3];
  const float* gamma = (const float*)d_in[4];
  const float* beta  = (const float*)d_in[5];

  const int N = in_sizes[0] / DF;
  const int E = in_sizes[1] / 2;
  const int* row = ei;
  const int* col = ei + E;

  // workspace layout (floats): h[N*DF] | agg[N*DF] | dinv[N] | stats[2*DF]
  float* h     = (float*)d_ws;
  float* agg   = h   + (size_t)N * DF;
  float* dinv  = agg + (size_t)N * DF;
  float* stats = dinv + N;

  k_init_deg  <<<(N + 255) / 256, 256, 0, stream>>>(dinv, N);
  k_count_deg <<<(E + 255) / 256, 256, 0, stream>>>(col, dinv, E);
  k_deg_to_dinv<<<(N + 255) / 256, 256, 0, stream>>>(dinv, N);

  k_gemm_wmma <<<(N + 15) / 16, 128, 0, stream>>>(x, W, h, N);

  k_agg_init  <<<N, 256, 0, stream>>>(h, dinv, bias, agg, N);
  k_scatter   <<<2048, 256, 0, stream>>>(row, col, dinv, h, agg, E);

  k_zero_stats<<<1, 2 * DF, 0, stream>>>(stats);
  k_stats     <<<(N + 127) / 128, 256, 0, stream>>>(agg, stats, N);
  k_bn_relu   <<<N, 256, 0, stream>>>(agg, stats, gamma, beta, (float*)d_out, N);
}